// XBMLoss_38062000177556
// MI455X (gfx1250) — compile-verified
//
#include <hip/hip_runtime.h>

typedef float v2f __attribute__((ext_vector_type(2)));
typedef float v8f __attribute__((ext_vector_type(8)));

#define DIM       128
#define KSTEPS    (DIM / 4)     // 32 k-steps of V_WMMA_F32_16X16X4_F32
#define NT_PER_WAVE 8           // n-tiles swept per wave (A reuse)
#define WAVES_PER_BLOCK 8       // 256 threads / wave32

__global__ void xbm_init_out(float* out) { out[0] = 0.0f; }

__global__ __launch_bounds__(256)
void xbm_loss_wmma_f32(const float* __restrict__ emb,
                       const int*   __restrict__ labels,
                       const float* __restrict__ ref,
                       const int*   __restrict__ ref_labels,
                       float*       __restrict__ out,
                       int lab_stride, float inv_n)
{
    const int lane = threadIdx.x & 31;
    const int wave = threadIdx.x >> 5;
    const int hi   = lane >> 4;       // 0: K pair {0,1}; 1: K pair {2,3}
    const int lo   = lane & 15;

    const int mBase  = blockIdx.x * 16;                          // m-tile
    const int gw     = blockIdx.y * WAVES_PER_BLOCK + wave;      // global wave in N
    const int nTile0 = gw * NT_PER_WAVE;

    // ---- A strip: 16 rows x 128 K, resident in 64 VGPRs (v2f per k-step) ----
    const float* arow = emb + (size_t)(mBase + lo) * DIM;
    v2f a[KSTEPS];
#pragma unroll
    for (int s = 0; s < KSTEPS; ++s)
        a[s] = *(const v2f*)(arow + 4 * s + 2 * hi);

    // m-labels for the 8 accumulator rows this lane observes (M = v + 8*hi)
    int labm[8];
#pragma unroll
    for (int v = 0; v < 8; ++v)
        labm[v] = labels[(size_t)(mBase + v + 8 * hi) * lab_stride];

    float lsum = 0.0f;

    for (int t = 0; t < NT_PER_WAVE; ++t) {
        const int    nBase = (nTile0 + t) * 16;
        const float* brow  = ref + (size_t)(nBase + lo) * DIM;
        const int    rlab  = ref_labels[(size_t)(nBase + lo) * lab_stride];

        v8f acc = {0.f, 0.f, 0.f, 0.f, 0.f, 0.f, 0.f, 0.f};
#pragma unroll
        for (int s = 0; s < KSTEPS; ++s) {
            v2f b = *(const v2f*)(brow + 4 * s + 2 * hi);
            // D = A x B + C, full fp32: v_wmma_f32_16x16x4_f32
            acc = __builtin_amdgcn_wmma_f32_16x16x4_f32(
                /*neg_a=*/false, a[s], /*neg_b=*/false, b,
                /*c_mod=*/(short)0, acc, /*reuse_a=*/false, /*reuse_b=*/false);
        }

        // Fused masked-loss epilogue on the 16x16 tile (no sim materialization)
#pragma unroll
        for (int v = 0; v < 8; ++v) {
            float sim  = acc[v];
            bool  same = (labm[v] == rlab);
            float c;
            if (same)
                c = (sim < 1.0f - 1e-5f) ? (1.0f - sim) : 0.0f;
            else
                c = (sim > 0.5f) ? sim : 0.0f;
            lsum += c;
        }
    }

    // wave32 reduction
#pragma unroll
    for (int off = 16; off >= 1; off >>= 1)
        lsum += __shfl_xor(lsum, off, 32);
    if (lane == 0)
        atomicAdd(out, lsum * inv_n);
}

extern "C" void kernel_launch(void* const* d_in, const int* in_sizes, int n_in,
                              void* d_out, int out_size, void* d_ws, size_t ws_size,
                              hipStream_t stream) {
    const float* emb        = (const float*)d_in[0];
    const int*   labels     = (const int*)  d_in[1];
    const float* ref        = (const float*)d_in[2];
    const int*   ref_labels = (const int*)  d_in[3];
    float*       out        = (float*)d_out;

    const int n = in_sizes[0] / DIM;             // 4096 query rows
    const int m = in_sizes[2] / DIM;             // 32768 reference rows
    const int lab_stride = in_sizes[1] / n;      // 2 (hierarchy levels)

    xbm_init_out<<<1, 1, 0, stream>>>(out);

    dim3 grid(n / 16, m / (16 * NT_PER_WAVE * WAVES_PER_BLOCK));  // 256 x 32
    xbm_loss_wmma_f32<<<grid, 32 * WAVES_PER_BLOCK, 0, stream>>>(
        emb, labels, ref, ref_labels, out, lab_stride, 1.0f / (float)n);
}